// ModelNew_3556232922418
// MI455X (gfx1250) — compile-verified
//
#include <hip/hip_runtime.h>

// Problem constants (fixed by the reference).
#define M_DIM 8192
#define N_DIM 4096
#define K_DIM 2048
#define EPSV  1e-5f

// Tiling.
#define BM 128          // block tile M
#define BN 64           // block tile N
#define KC 32           // K chunk staged in LDS (dwords per row)
#define PAD 4           // TDM-inserted pad dwords per row
#define LDSTR (KC+PAD)  // 36: conflict-free, 8B-aligned b64 fragment reads

typedef __attribute__((ext_vector_type(2))) float v2f;
typedef __attribute__((ext_vector_type(4))) float f4v;
typedef __attribute__((ext_vector_type(8))) float v8f;
typedef __attribute__((ext_vector_type(4))) unsigned int u32x4;
typedef __attribute__((ext_vector_type(4))) int i32x4;
typedef __attribute__((ext_vector_type(8))) int i32x8;

__device__ __forceinline__ v8f wmma4(v2f a, v2f b, v8f c) {
    // D = A(16x4, f32) * B(4x16, f32) + C(16x16, f32)
    return __builtin_amdgcn_wmma_f32_16x16x4_f32(
        /*neg_a=*/false, a, /*neg_b=*/false, b,
        /*c_mod=*/(short)0, c, /*reuse_a=*/false, /*reuse_b=*/false);
}

__device__ __forceinline__ float sxor(float v, int m) { return __shfl_xor(v, m, 32); }

// Float atomic-min via sign-aware integer atomics (monotone bit trick).
// Valid because rowmin[] is initialized to +inf (positive bit pattern).
__device__ __forceinline__ void atomicMinF(float* a, float v) {
    if (v >= 0.0f) atomicMin((int*)a,          __float_as_int(v));
    else           atomicMax((unsigned int*)a, __float_as_uint(v));
}

// Group-norm normalize: group = 8 consecutive N-columns = 8 consecutive lanes
// in the WMMA C layout. Returns (v - mean) * rsqrt(var + eps).
__device__ __forceinline__ float gnorm(float v) {
    float s = v;
    s += sxor(s, 1); s += sxor(s, 2); s += sxor(s, 4);
    float mean = s * 0.125f;
    float d = v - mean;
    float q = d * d;
    q += sxor(q, 1); q += sxor(q, 2); q += sxor(q, 4);
    return d * rsqrtf(q * 0.125f + EPSV);
}

// ---------------------------------------------------------------------------
// Tensor Data Mover: async DMA of a 2D tile (tile_rows x KC dwords) from a
// row-major f32 tensor (tensor_rows x K_DIM) into LDS, with hardware padding
// of PAD dwords after every KC dwords (CDNA5 ISA).
//   group0: [1:0]=count=1 | [63:32]=lds_addr | [120:64]=global_addr | [127:126]=type=2
//   group1: data_size=4B(2), pad_enable, pad_interval=32dw(4), pad_amount=4dw(3),
//           tensor_dim0=K_DIM, tensor_dim1=tensor_rows, tile_dim0=KC,
//           tile_dim1=tile_rows, tensor_dim0_stride=K_DIM
//   groups 2/3: zero (2-D tensor)
// ---------------------------------------------------------------------------
__device__ __forceinline__ void tdm_load_tile(const float* gptr, void* lds_ptr,
                                              int tensor_rows, int tile_rows) {
    unsigned long long ga = (unsigned long long)gptr;
    unsigned lds_off = (unsigned)(unsigned long long)lds_ptr;   // AS(3) offset

    u32x4 g0 = { 1u,                                   // count=1, user mode
                 lds_off,                              // lds_addr
                 (unsigned)ga,                         // global_addr[31:0]
                 (unsigned)(ga >> 32) | (2u << 30) };  // global_addr[56:32] | type=2

    int d0 = (2 << 16) | (1 << 20) | (4 << 22) | (3 << 25); // 4B elems, pad 32dw->+4dw
    int d1 = (K_DIM & 0xffff) << 16;                        // tensor_dim0 lo
    int d2 = (K_DIM >> 16) | ((tensor_rows & 0xffff) << 16);// tensor_dim0 hi | dim1 lo
    int d3 = (tensor_rows >> 16) | (KC << 16);              // dim1 hi | tile_dim0
    int d4 = tile_rows;                                     // tile_dim1 (tile_dim2=0)
    int d5 = K_DIM;                                         // tensor_dim0_stride lo
    i32x8 g1 = { d0, d1, d2, d3, d4, d5, 0, 0 };
    i32x4 gz = { 0, 0, 0, 0 };

#if __has_include(<hip/amd_detail/amd_gfx1250_TDM.h>)
    i32x8 gz8 = { 0, 0, 0, 0, 0, 0, 0, 0 };
    __builtin_amdgcn_tensor_load_to_lds(g0, g1, gz, gz, gz8, 0);  // clang-23 form
#else
    __builtin_amdgcn_tensor_load_to_lds(g0, g1, gz, gz, 0);       // ROCm 7.2 form
#endif
}

__global__ __launch_bounds__(256) void init_rowmin_kernel(float* __restrict__ rm) {
    rm[blockIdx.x * 256 + threadIdx.x] = __int_as_float(0x7F800000); // +inf
}

// Fused GEMM (+b) -> group-norm -> row-min kernel, TDM-staged double buffer.
// grid = (M/BM, N/BN), block = 256 (8 waves; 4 along M x 2 along N; 32x32/wave).
__global__ __launch_bounds__(256) void gemm_gn_min_kernel(
        const float* __restrict__ x,   // (M, K)
        const float* __restrict__ W,   // (N, K)
        const float* __restrict__ b,   // (N)
        float* __restrict__ rowmin)    // (M), pre-set to +inf
{
    __shared__ float As[2][BM * LDSTR];
    __shared__ float Bs[2][BN * LDSTR];

    const int tid  = threadIdx.x;
    const int wave = tid >> 5;
    const int lane = tid & 31;
    const int half = lane >> 4;   // 0: K,K+1 / C rows M..M+7; 1: K+2,K+3 / rows +8
    const int lrow = lane & 15;
    const int wm   = wave & 3;    // wave tile row (x32)
    const int wn   = wave >> 2;   // wave tile col (x32)

    const int mbase = blockIdx.x * BM;
    const int nbase = blockIdx.y * BN;

    v8f c00 = {}, c01 = {}, c10 = {}, c11 = {};

    // Prologue: wave 0 kicks off the TDM for chunk 0.
    if (wave == 0) {
        tdm_load_tile(x + (size_t)mbase * K_DIM, &As[0][0], M_DIM, BM);
        tdm_load_tile(W + (size_t)nbase * K_DIM, &Bs[0][0], N_DIM, BN);
    }

    int buf = 0;
    for (int kb = 0; kb < K_DIM; kb += KC) {
        // Issue next chunk's DMA into the other buffer (one chunk in flight).
        if (kb + KC < K_DIM) {
            if (wave == 0) {
                tdm_load_tile(x + (size_t)mbase * K_DIM + kb + KC, &As[buf ^ 1][0], M_DIM, BM);
                tdm_load_tile(W + (size_t)nbase * K_DIM + kb + KC, &Bs[buf ^ 1][0], N_DIM, BN);
            }
            __builtin_amdgcn_s_wait_tensorcnt(2);   // current chunk's 2 loads done (FIFO)
        } else {
            __builtin_amdgcn_s_wait_tensorcnt(0);
        }
        __syncthreads();   // publish LDS tile to all waves

        // Fragment base pointers: A 16x4 / B 4x16 fragments are per-lane
        // float2 loads (lane<16: k+0..1, lane>=16: k+2..3).
        const float* A0 = &As[buf][0] + (wm * 32 +      lrow) * LDSTR + half * 2;
        const float* A1 = &As[buf][0] + (wm * 32 + 16 + lrow) * LDSTR + half * 2;
        const float* B0 = &Bs[buf][0] + (wn * 32 +      lrow) * LDSTR + half * 2;
        const float* B1 = &Bs[buf][0] + (wn * 32 + 16 + lrow) * LDSTR + half * 2;

        #pragma unroll
        for (int kk = 0; kk < KC; kk += 4) {
            v2f a0 = *(const v2f*)(A0 + kk);
            v2f a1 = *(const v2f*)(A1 + kk);
            v2f b0 = *(const v2f*)(B0 + kk);
            v2f b1 = *(const v2f*)(B1 + kk);
            c00 = wmma4(a0, b0, c00);
            c01 = wmma4(a0, b1, c01);
            c10 = wmma4(a1, b0, c10);
            c11 = wmma4(a1, b1, c11);
        }
        __syncthreads();   // all reads done before TDM overwrites this buffer
        buf ^= 1;
    }

    // Epilogue: +b, group-norm (groups of 8 cols), row-min over this wave's
    // 32 columns, one atomic per row per 32-col tile.
    const int cb = nbase + wn * 32 + lrow;   // column owned by this lane (both halves)
    const float bv0 = b[cb];
    const float bv1 = b[cb + 16];
    const int r0 = mbase + wm * 32 + half * 8;

    #pragma unroll
    for (int i = 0; i < 8; ++i) {
        float n0 = gnorm(c00[i] + bv0);
        float n1 = gnorm(c01[i] + bv1);
        float mn = fminf(n0, n1);
        mn = fminf(mn, sxor(mn, 1));
        mn = fminf(mn, sxor(mn, 2));
        mn = fminf(mn, sxor(mn, 4));
        mn = fminf(mn, sxor(mn, 8));
        if (lrow == 0) atomicMinF(&rowmin[r0 + i], mn);

        float m0 = gnorm(c10[i] + bv0);
        float m1 = gnorm(c11[i] + bv1);
        float mn2 = fminf(m0, m1);
        mn2 = fminf(mn2, sxor(mn2, 1));
        mn2 = fminf(mn2, sxor(mn2, 2));
        mn2 = fminf(mn2, sxor(mn2, 4));
        mn2 = fminf(mn2, sxor(mn2, 8));
        if (lrow == 0) atomicMinF(&rowmin[r0 + 16 + i], mn2);
    }
}

// out[m, n] = rowmin[m] + bias[n]; vectorized float4 streaming store (128 MB).
__global__ __launch_bounds__(256) void broadcast_kernel(
        const float* __restrict__ rowmin,
        const float* __restrict__ bias,
        float* __restrict__ out)
{
    size_t idx = (size_t)blockIdx.x * 256 + threadIdx.x;   // over M * (N/4)
    int m  = (int)(idx >> 10);                              // N/4 = 1024
    int nq = (int)(idx & 1023);
    float mn = rowmin[m];
    f4v bb = *(const f4v*)(bias + nq * 4);
    f4v o = bb + mn;
    *(f4v*)(out + ((size_t)m << 12) + (size_t)nq * 4) = o;
}

extern "C" void kernel_launch(void* const* d_in, const int* in_sizes, int n_in,
                              void* d_out, int out_size, void* d_ws, size_t ws_size,
                              hipStream_t stream) {
    (void)in_sizes; (void)n_in; (void)out_size; (void)ws_size;
    const float* x    = (const float*)d_in[0];   // (8192, 2048)
    const float* W    = (const float*)d_in[1];   // (4096, 2048)
    const float* b    = (const float*)d_in[2];   // (4096)
    const float* bias = (const float*)d_in[3];   // (4096)
    float* out    = (float*)d_out;               // (8192*4096)
    float* rowmin = (float*)d_ws;                // (8192) scratch

    // 1) rowmin = +inf (must re-init every call; ws is not re-poisoned).
    init_rowmin_kernel<<<M_DIM / 256, 256, 0, stream>>>(rowmin);

    // 2) fused GEMM + bias + group-norm + row-min (TDM-staged WMMA).
    dim3 grid(M_DIM / BM, N_DIM / BN);
    gemm_gn_min_kernel<<<grid, 256, 0, stream>>>(x, W, b, rowmin);

    // 3) out = rowmin[m] + bias[n].
    size_t total = (size_t)M_DIM * (N_DIM / 4);
    broadcast_kernel<<<(unsigned)(total / 256), 256, 0, stream>>>(rowmin, bias, out);
}